// Model_83141976916201
// MI455X (gfx1250) — compile-verified
//
#include <hip/hip_runtime.h>
#include <hip/hip_bf16.h>

// ---- problem constants (match reference) ----
#define B_SZ   16
#define SEQLEN 2048
#define NH     16
#define PDIM   64
#define NDIM   16
#define LBLK   64
#define NCHUNK (SEQLEN / LBLK)

// ---- LDS strides, padded for bank-conflict-free WMMA operand loads ----
#define LDX 68   // 64x64 X / M buffers: bank = (4*row + col) % 64 -> conflict free
#define LDB 17   // 64x16 B / C / S buffers: bank = (17*row + col) % 64 -> conflict free

#if __has_builtin(__builtin_amdgcn_global_load_async_to_lds_b128)
#define HAS_ASYNC_LDS 1
#else
#define HAS_ASYNC_LDS 0
#endif

typedef __attribute__((ext_vector_type(2))) float v2f;
typedef __attribute__((ext_vector_type(8))) float v8f;

#if HAS_ASYNC_LDS
// builtin parameter types per hipcc diagnostic:
//   arg0: int __attribute__((vector_size(16))) addrspace(1)*   (global src)
//   arg1: int __attribute__((vector_size(16))) addrspace(3)*   (LDS dst)
typedef int b128_t __attribute__((vector_size(16)));
typedef __attribute__((address_space(1))) b128_t* gas_b128_ptr;
typedef __attribute__((address_space(3))) b128_t* lds_b128_ptr;
#endif

__device__ __forceinline__ void wait_asynccnt0() {
#if __has_builtin(__builtin_amdgcn_s_wait_asynccnt)
  __builtin_amdgcn_s_wait_asynccnt(0);
#else
  asm volatile("s_wait_asynccnt 0x0" ::: "memory");
#endif
}

// D = A(16x4) * B(4x16) + C, fp32 WMMA (wave32)
__device__ __forceinline__ v8f wmma4(v2f a, v2f b, v8f c) {
  return __builtin_amdgcn_wmma_f32_16x16x4_f32(
      /*neg_a=*/false, a, /*neg_b=*/false, b,
      /*c_mod=*/(short)0, c, /*reuse_a=*/false, /*reuse_b=*/false);
}

// A-operand tile load (16x4) from row-major LDS buffer, tile origin (row0,k0).
// ISA layout: lane holds row m=lane&15; lane-half selects k pair {2h, 2h+1}.
__device__ __forceinline__ v2f load_a(const float* buf, int ld, int row0, int k0) {
  const int lane = threadIdx.x & 31;
  const int m  = lane & 15;
  const int kk = (lane >> 4) << 1;
  const float* p = buf + (row0 + m) * ld + k0 + kk;
  v2f r; r[0] = p[0]; r[1] = p[1];
  return r;
}

// B-operand tile load (4x16) from row-major (K-major rows) LDS buffer.
// ISA layout: lane holds col n=lane&15; lane-half selects rows {2h, 2h+1}.
__device__ __forceinline__ v2f load_b(const float* buf, int ld, int k0, int col0) {
  const int lane = threadIdx.x & 31;
  const int n  = lane & 15;
  const int kk = (lane >> 4) << 1;
  const float* p = buf + (k0 + kk) * ld + col0 + n;
  v2f r; r[0] = p[0]; r[1] = p[ld];
  return r;
}

// B-operand tile load (4x16) of the TRANSPOSE of a row-major NxK buffer:
// element [k][n] = buf[(col0+n)*ld + k0 + k]  (k-pair contiguous within a row).
__device__ __forceinline__ v2f load_bT(const float* buf, int ld, int k0, int col0) {
  const int lane = threadIdx.x & 31;
  const int n  = lane & 15;
  const int kk = (lane >> 4) << 1;
  const float* p = buf + (col0 + n) * ld + k0 + kk;
  v2f r; r[0] = p[0]; r[1] = p[1];
  return r;
}

__global__ __launch_bounds__(256) void ssd_fused_kernel(
    const float* __restrict__ Xg, const float* __restrict__ Ag,
    const float* __restrict__ Bg, const float* __restrict__ Cg,
    float* __restrict__ Yg) {
  __shared__ __align__(16) float Xs[LBLK * LDX];   // X chunk (l x p)
  __shared__ __align__(16) float Ms[LBLK * LDX];   // CB ⊙ L   (l x s)
  __shared__ __align__(16) float Bs[LBLK * LDB];   // B chunk (l x n)
  __shared__ __align__(16) float Cs[LBLK * LDB];   // C chunk (l x n)
  __shared__ __align__(16) float Ss[PDIM * LDB];   // carried state S_in (p x n)
  __shared__ float Acs[LBLK], expA[LBLK], rexpA[LBLK], decayS[LBLK];

  const int b    = blockIdx.x >> 4;
  const int h    = blockIdx.x & 15;
  const int tid  = threadIdx.x;
  const int wave = tid >> 5;
  const int lane = tid & 31;
  const int lt   = wave >> 1;   // row strip (l or p), 0..3
  const int cg   = wave & 1;    // column-tile group {0,1} vs {2,3}
  const int nidx = lane & 15;
  const int half = lane >> 4;

  // zero carried state
  for (int i = tid; i < PDIM * LDB; i += 256) Ss[i] = 0.0f;
  __syncthreads();

  for (int ci = 0; ci < NCHUNK; ++ci) {
    const int s0 = ci * LBLK;

    // ---- stage A, X, B, C into LDS ----
    if (tid < LBLK)
      Acs[tid] = Ag[((size_t)b * SEQLEN + s0 + tid) * NH + h];
#if HAS_ASYNC_LDS
    // CDNA5 async DMA path: global -> LDS without VGPR round trip (ASYNCcnt)
    for (int i = tid; i < LBLK * 16; i += 256) {      // X: 64 rows x 16 b128
      const int r = i >> 4, c4 = i & 15;
      const float* gp = Xg + ((((size_t)b * SEQLEN + s0 + r) * NH) + h) * PDIM + c4 * 4;
      float* lp = &Xs[r * LDX + c4 * 4];              // LDX%4==0 -> 16B aligned
      __builtin_amdgcn_global_load_async_to_lds_b128(
          (gas_b128_ptr)(void*)const_cast<float*>(gp),
          (lds_b128_ptr)(void*)lp, 0, 0);
    }
#else
    for (int i = tid; i < LBLK * 16; i += 256) {      // X: 64 rows x 16 float4
      const int r = i >> 4, c4 = i & 15;
      const float4 v = *(const float4*)(Xg + ((((size_t)b * SEQLEN + s0 + r) * NH) + h) * PDIM + c4 * 4);
      *(float4*)(&Xs[r * LDX + c4 * 4]) = v;
    }
#endif
    for (int i = tid; i < LBLK * 4; i += 256) {       // B,C: 64 rows x 4 float4
      const int r = i >> 2, c4 = i & 3;
      const size_t g = ((((size_t)b * SEQLEN + s0 + r) * NH) + h) * NDIM + c4 * 4;
      const float4 vb = *(const float4*)(Bg + g);
      const float4 vc = *(const float4*)(Cg + g);
      float* db = &Bs[r * LDB + c4 * 4];
      float* dc = &Cs[r * LDB + c4 * 4];
      db[0] = vb.x; db[1] = vb.y; db[2] = vb.z; db[3] = vb.w;
      dc[0] = vc.x; dc[1] = vc.y; dc[2] = vc.z; dc[3] = vc.w;
    }
    // prefetch next chunk's X rows while we compute (global_prefetch_b8)
    if (ci + 1 < NCHUNK && tid < LBLK)
      __builtin_prefetch(Xg + ((((size_t)b * SEQLEN + s0 + LBLK + tid) * NH) + h) * PDIM, 0, 1);
#if HAS_ASYNC_LDS
    wait_asynccnt0();   // per-wave: my async transfers landed in LDS
#endif
    __syncthreads();

    if (tid == 0) {                                   // inclusive cumsum of A
      float s = 0.0f;
      for (int i = 0; i < LBLK; ++i) { s += Acs[i]; Acs[i] = s; }
    }
    __syncthreads();
    if (tid < LBLK) {
      const float v = Acs[tid];
      expA[tid]   = __expf(v);
      rexpA[tid]  = __expf(-v);
      decayS[tid] = __expf(Acs[LBLK - 1] - v);
    }
    __syncthreads();

    // hoist per-lane row scales out of the tile loops (kills the branchy
    // exec-mask codegen around per-element LDS loads in the mask epilogue)
    float eAr[8];
    #pragma unroll
    for (int r = 0; r < 8; ++r) eAr[r] = expA[lt * 16 + r + 8 * half];
    const float eAm    = expA[lt * 16 + nidx];  // Y_off A-operand row scale
    const float eaLast = expA[LBLK - 1];        // inter-chunk decay

    // ---- M = (C · Bᵀ) ⊙ L, lower-triangular tiles only ----
    for (int stI = 0; stI < 2; ++stI) {
      const int st = 2 * cg + stI;
      if (st > lt) continue;
      v8f acc = {};
      #pragma unroll
      for (int kt = 0; kt < 4; ++kt) {
        const v2f a  = load_a (Cs, LDB, lt * 16, kt * 4);
        const v2f bb = load_bT(Bs, LDB, kt * 4, st * 16);  // Bᵀ[k=n][col=s]
        acc = wmma4(a, bb, acc);
      }
      const int scol = st * 16 + nidx;
      const float rs = rexpA[scol];
      #pragma unroll
      for (int r = 0; r < 8; ++r) {
        const int lrow = lt * 16 + r + 8 * half;
        const float v = acc[r] * eAr[r] * rs;
        Ms[lrow * LDX + scol] = (scol <= lrow) ? v : 0.0f;   // v_cndmask, no branch
      }
    }
    __syncthreads();

    // ---- Y = M · X  +  (expA ⊙ C) · S_inᵀ ----
    for (int ptI = 0; ptI < 2; ++ptI) {
      const int pt = 2 * cg + ptI;
      v8f acc = {};
      const int ksteps = (lt + 1) * 4;                // M is zero past the diagonal
      for (int kt = 0; kt < ksteps; ++kt) {
        const v2f a  = load_a(Ms, LDX, lt * 16, kt * 4);
        const v2f bb = load_b(Xs, LDX, kt * 4, pt * 16);
        acc = wmma4(a, bb, acc);
      }
      #pragma unroll
      for (int kt = 0; kt < 4; ++kt) {                // K = n = 16
        v2f a = load_a(Cs, LDB, lt * 16, kt * 4);
        a[0] *= eAm; a[1] *= eAm;                     // row scale exp(A_cs[l])
        const v2f bb = load_bT(Ss, LDB, kt * 4, pt * 16);  // S_inᵀ[k=n][col=p]
        acc = wmma4(a, bb, acc);
      }
      const int pcol = pt * 16 + nidx;
      #pragma unroll
      for (int r = 0; r < 8; ++r) {
        const int lrow = lt * 16 + r + 8 * half;
        Yg[((((size_t)b * SEQLEN) + s0 + lrow) * NH + h) * PDIM + pcol] = acc[r];
      }
    }

    // ---- chunk state = (decay ⊙ X)ᵀ · B  (p x n), even waves ----
    v8f sacc = {};
    if (cg == 0) {
      #pragma unroll
      for (int kt = 0; kt < 16; ++kt) {
        const int kk = (lane >> 4) << 1;
        const int l0 = kt * 4 + kk;
        v2f a;                                        // A[m=p][k=l] = decay[l]*X[l][p]
        a[0] = decayS[l0]     * Xs[(l0)     * LDX + lt * 16 + nidx];
        a[1] = decayS[l0 + 1] * Xs[(l0 + 1) * LDX + lt * 16 + nidx];
        const v2f bb = load_b(Bs, LDB, kt * 4, 0);    // B[k=l][n]
        sacc = wmma4(a, bb, sacc);
      }
    }
    __syncthreads();   // everyone done reading old S_in / X / B / C

    // ---- inter-chunk scan: S_in = exp(A_cs[last]) * S_in + chunk_state ----
    if (cg == 0) {
      #pragma unroll
      for (int r = 0; r < 8; ++r) {
        const int p = lt * 16 + r + 8 * half;
        Ss[p * LDB + nidx] = eaLast * Ss[p * LDB + nidx] + sacc[r];
      }
    }
    __syncthreads();
  }
}

extern "C" void kernel_launch(void* const* d_in, const int* in_sizes, int n_in,
                              void* d_out, int out_size, void* d_ws, size_t ws_size,
                              hipStream_t stream) {
  (void)in_sizes; (void)n_in; (void)d_ws; (void)ws_size; (void)out_size;
  const float* X = (const float*)d_in[0];
  const float* A = (const float*)d_in[1];
  const float* B = (const float*)d_in[2];
  const float* C = (const float*)d_in[3];
  float* Y = (float*)d_out;
  dim3 grid(B_SZ * NH);     // one persistent workgroup per (b, h)
  dim3 block(256);          // 8 wave32 waves
  ssd_fused_kernel<<<grid, block, 0, stream>>>(X, A, B, C, Y);
}